// PNTripletloss_60361470378721
// MI455X (gfx1250) — compile-verified
//
#include <hip/hip_runtime.h>
#include <hip/hip_bf16.h>

// ---------------------------------------------------------------------------
// PNTriplet loss, MI455X (gfx1250, wave32).
// Memory-bound streaming reduction: 256 MB in / 4 B out -> ~11us @ 23.3 TB/s.
// Strategy: b128 nontemporal grid-stride loads, fused per-element math,
// wave-level reduction via one V_WMMA_F32_16X16X4_F32 (B = ones), then a
// deterministic two-kernel tree reduction (no atomics, no d_out pre-init).
// ---------------------------------------------------------------------------

typedef __attribute__((ext_vector_type(2))) float     v2f;
typedef __attribute__((ext_vector_type(4))) float     v4f;
typedef __attribute__((ext_vector_type(8))) float     v8f;
typedef __attribute__((ext_vector_type(2))) long long v2ll;

#define PNT_L_COEF 0.1f
#define PNT_MARGIN 5.0f
#define PNT_BLOCKS 2048
#define PNT_TPB    256

// Per-element fused term: (logsumexp - p_y) + 0.1 * relu(diff + 5)
__device__ __forceinline__ float pnt_term(float p0, float p1, long long lab) {
    const bool z   = (lab == 0);
    const float py = z ? p0 : p1;          // selected-class logit
    const float po = z ? p1 : p0;          // other-class logit
    const float mx = fmaxf(p0, p1);
    const float mn = fminf(p0, p1);
    // lse = max + log(1 + exp(min - max)); exp arg <= 0, 1+e in [1,2]
    const float lse = mx + __logf(1.0f + __expf(mn - mx));
    const float t1  = lse - py;                              // -log_softmax[y]
    const float t2  = fmaxf((po - py) + PNT_MARGIN, 0.0f);   // margin hinge
    return t1 + PNT_L_COEF * t2;
}

// Wave32 sum of 64 fp32 partials (2 per lane) using one matrix op.
// A (16x4 layout): lanes 0-15 carry rows M=lane with K=0,1 in VGPR0,1;
// lanes 16-31 carry K=2,3.  B = ones(4x16).  D[m][n] = rowsum(m).
// D layout: d[j] = rowsum(j) in lanes 0-15, rowsum(j+8) in lanes 16-31.
// => t = sum(d[0..7]) holds half-totals in lanes 0 and 16.
__device__ __forceinline__ float wave_sum_wmma(float a0, float a1) {
    v2f A; A[0] = a0;   A[1] = a1;
    v2f B; B[0] = 1.0f; B[1] = 1.0f;
    v8f C = {};
    v8f D = __builtin_amdgcn_wmma_f32_16x16x4_f32(
        /*neg_a=*/false, A, /*neg_b=*/false, B,
        /*c_mod=*/(short)0, C, /*reuse_a=*/false, /*reuse_b=*/false);
    float t = ((D[0] + D[1]) + (D[2] + D[3])) + ((D[4] + D[5]) + (D[6] + D[7]));
    return __shfl(t, 0, 32) + __shfl(t, 16, 32);   // total broadcast to all lanes
}

__device__ __forceinline__ float block_sum(float acc0, float acc1) {
    __shared__ float sdata[PNT_TPB / 32];
    const float wsum = wave_sum_wmma(acc0, acc1);  // EXEC all-ones here
    const int lane = threadIdx.x & 31;
    const int wid  = threadIdx.x >> 5;
    if (lane == 0) sdata[wid] = wsum;
    __syncthreads();
    float b = 0.0f;
    if (threadIdx.x == 0) {
        const int nw = blockDim.x >> 5;
        for (int i = 0; i < nw; ++i) b += sdata[i];
    }
    return b;  // valid in thread 0 only
}

// Pass 1: stream y_pred (float4 = 2 rows) + labels (2x int64) -> per-block sum.
__global__ void pnt_partial_kernel(const v4f* __restrict__ yp2,
                                   const v2ll* __restrict__ yl2,
                                   float* __restrict__ partial,
                                   long long n) {
    const long long nPairs  = n >> 1;
    const long long gstride = (long long)gridDim.x * blockDim.x;
    long long p = (long long)blockIdx.x * blockDim.x + threadIdx.x;

    float acc0 = 0.0f, acc1 = 0.0f;  // dual accumulators for ILP
    for (; p < nPairs; p += gstride) {
        // Non-temporal: 256 MB stream > 192 MB L2, do not cache.
        const v4f  v = __builtin_nontemporal_load(&yp2[p]);
        const v2ll l = __builtin_nontemporal_load(&yl2[p]);
        acc0 += pnt_term(v[0], v[1], l[0]);
        acc1 += pnt_term(v[2], v[3], l[1]);
    }
    // Odd-N tail (not hit for N=16M, kept for generality); resolves before WMMA.
    if ((n & 1LL) && blockIdx.x == 0 && threadIdx.x == 0) {
        const float* yp = (const float*)yp2;
        const long long* yl = (const long long*)yl2;
        acc0 += pnt_term(yp[2 * (n - 1)], yp[2 * (n - 1) + 1], yl[n - 1]);
    }

    const float bsum = block_sum(acc0, acc1);
    if (threadIdx.x == 0) partial[blockIdx.x] = bsum;
}

// Pass 2: single block reduces the per-block partials and scales by 1/N.
__global__ void pnt_final_kernel(const float* __restrict__ partial,
                                 int nPartials,
                                 float* __restrict__ out,
                                 float inv_n) {
    float acc0 = 0.0f, acc1 = 0.0f;
    for (int i = threadIdx.x; i < nPartials; i += 2 * blockDim.x) {
        acc0 += partial[i];
        const int j = i + blockDim.x;
        if (j < nPartials) acc1 += partial[j];
    }
    const float total = block_sum(acc0, acc1);
    if (threadIdx.x == 0) out[0] = total * inv_n;
}

extern "C" void kernel_launch(void* const* d_in, const int* in_sizes, int n_in,
                              void* d_out, int out_size, void* d_ws, size_t ws_size,
                              hipStream_t stream) {
    const v4f*  yp2 = (const v4f*)d_in[0];   // y_pred: (N,2) fp32 -> float4 = 2 rows
    const v2ll* yl2 = (const v2ll*)d_in[1];  // y: (N,) int64 -> 2 labels per b128
    const long long n = (long long)in_sizes[1];

    int blocks = PNT_BLOCKS;
    if (ws_size < (size_t)blocks * sizeof(float))
        blocks = (int)(ws_size / sizeof(float));
    if (blocks < 1) blocks = 1;

    float* partial = (float*)d_ws;
    pnt_partial_kernel<<<blocks, PNT_TPB, 0, stream>>>(yp2, yl2, partial, n);
    pnt_final_kernel<<<1, PNT_TPB, 0, stream>>>(partial, blocks, (float*)d_out,
                                                1.0f / (float)n);
}